// ParametricGCNEncoder_47167330844989
// MI455X (gfx1250) — compile-verified
//
#include <hip/hip_runtime.h>
#include <hip/hip_bf16.h>

typedef __attribute__((ext_vector_type(2))) float v2f;
typedef __attribute__((ext_vector_type(8))) float v8f;

#define F_IN 64
#define HDIM 128
#define ODIM 64

// ---------- utility kernels ----------

__global__ void init_const_kernel(float* __restrict__ p, float v, int n) {
    int i = blockIdx.x * blockDim.x + threadIdx.x;
    if (i < n) p[i] = v;
}

// deg[dst] += 1 per edge (deg pre-initialized to 1.0 for the self-loop)
__global__ void degree_kernel(const int* __restrict__ dst, float* __restrict__ deg, int E) {
    int e = blockIdx.x * blockDim.x + threadIdx.x;
    if (e < E) atomicAdd(&deg[dst[e]], 1.0f);
}

// deg -> deg^-0.5 in place
__global__ void dinv_kernel(float* __restrict__ deg, int n) {
    int i = blockIdx.x * blockDim.x + threadIdx.x;
    if (i < n) {
        float d = deg[i];
        deg[i] = (d > 0.0f) ? rsqrtf(d) : 0.0f;
    }
}

// ---------- WMMA f32 GEMM: C[M,Nn] = A[M,K] @ B[K,Nn] ----------
// K, Nn compile-time -> fully unrolled K loop, immediate-offset loads.
// One wave computes a 16x32 output slab (two 16x16 tiles sharing the A fragment).
// V_WMMA_F32_16X16X4_F32 lane layout:
//   A 16x4 : lanes 0-15 rows 0..15 K={k0,k0+1}; lanes 16-31 K={k0+2,k0+3}
//   B 4x16 : VGPR0 lanes 0-15 -> K=k0 N=lane, lanes 16-31 -> K=k0+2; VGPR1 -> k0+1/k0+3
//   C/D    : VGPR v, lanes 0-15 -> M=v N=lane; lanes 16-31 -> M=v+8
template <int K, int Nn>
__global__ void gemm_wmma_kernel(const float* __restrict__ A, const float* __restrict__ B,
                                 float* __restrict__ C, int M) {
    constexpr int NP = Nn / 32;                 // n-tile pairs
    int lane = threadIdx.x & 31;
    int wave = (blockIdx.x * blockDim.x + threadIdx.x) >> 5;
    int ntm = (M + 15) >> 4;
    if (wave >= ntm * NP) return;               // wave-uniform exit: EXEC all-1s inside
    int tm = wave / NP;
    int tp = wave - tm * NP;

    int half = lane >> 4;                       // 0: K pair lo, 1: K pair hi
    int l    = lane & 15;
    int arow = tm * 16 + l;
    int arow_c = (arow < M) ? arow : 0;         // clamp: garbage rows are never stored
    const float* ap = A + (size_t)arow_c * K + (half << 1);
    const float* bp = B + (size_t)(half << 1) * Nn + tp * 32 + l;

    v8f acc0 = {0.f, 0.f, 0.f, 0.f, 0.f, 0.f, 0.f, 0.f};
    v8f acc1 = {0.f, 0.f, 0.f, 0.f, 0.f, 0.f, 0.f, 0.f};

#pragma unroll
    for (int k0 = 0; k0 < K; k0 += 4) {
        v2f a = *(const v2f*)(ap + k0);         // 8B contiguous, immediate offset
        v2f b0, b1;
        b0.x = bp[k0 * Nn];
        b0.y = bp[(k0 + 1) * Nn];
        b1.x = bp[k0 * Nn + 16];
        b1.y = bp[(k0 + 1) * Nn + 16];
        acc0 = __builtin_amdgcn_wmma_f32_16x16x4_f32(false, a, false, b0, (short)0, acc0,
                                                     false, false);
        acc1 = __builtin_amdgcn_wmma_f32_16x16x4_f32(false, a, false, b1, (short)0, acc1,
                                                     false, false);
    }

    int rbase = tm * 16 + half * 8;
    float* cp = C + (size_t)rbase * Nn + tp * 32 + l;
#pragma unroll
    for (int v = 0; v < 8; ++v) {
        if (rbase + v < M) {
            cp[v * Nn] = acc0[v];
            cp[v * Nn + 16] = acc1[v];
        }
    }
}

// ---------- GCN aggregation ----------

// agg[i,f] = h[i,f] * dinv[i]^2 + bias[f]   (self-loop term + bias, also inits agg)
__global__ void selfloop_bias_kernel(const float* __restrict__ h, const float* __restrict__ dinv,
                                     const float* __restrict__ bias, float* __restrict__ agg,
                                     int Nn, int logC) {
    long long i = blockIdx.x * (long long)blockDim.x + threadIdx.x;
    long long total = (long long)Nn << logC;
    if (i >= total) return;
    int node = (int)(i >> logC);
    int f = (int)(i & ((1 << logC) - 1));
    float di = dinv[node];
    agg[i] = h[i] * di * di + bias[f];
}

// per edge: agg[dst] += h[src] * dinv[src]*dinv[dst]; one lane handles 4 features
// logPE = log2(C/4): 32 lanes cover C=128 (one edge per wave), 16 lanes cover C=64
__global__ void scatter_kernel(const float* __restrict__ h, const int* __restrict__ src,
                               const int* __restrict__ dst, const float* __restrict__ dinv,
                               float* __restrict__ agg, int E, int C, int logPE) {
    long long tid = blockIdx.x * (long long)blockDim.x + threadIdx.x;
    long long e = tid >> logPE;
    int f4 = (int)(tid & ((1 << logPE) - 1)) << 2;
    if (e >= E) return;
    int s = src[e];
    int d = dst[e];
    float w = dinv[s] * dinv[d];
    const float4 hv = *(const float4*)(h + (long long)s * C + f4);
    float* out = agg + (long long)d * C + f4;
    atomicAdd(out + 0, hv.x * w);
    atomicAdd(out + 1, hv.y * w);
    atomicAdd(out + 2, hv.z * w);
    atomicAdd(out + 3, hv.w * w);
}

__global__ void relu_kernel(float* __restrict__ p, long long n) {
    long long i = blockIdx.x * (long long)blockDim.x + threadIdx.x;
    if (i < n) p[i] = fmaxf(p[i], 0.0f);
}

// ---------- pooling ----------

__global__ void pool_sum_kernel(const float* __restrict__ agg, const int* __restrict__ batch,
                                float* __restrict__ psum, int Nn) {
    int tid = blockIdx.x * blockDim.x + threadIdx.x;   // Nn * (ODIM/4) threads
    int node = tid >> 4;                                // ODIM/4 == 16 lanes per node
    int f4 = (tid & 15) << 2;
    if (node >= Nn) return;
    int b = batch[node];
    const float4 v = *(const float4*)(agg + (long long)node * ODIM + f4);
    float* o = psum + (long long)b * ODIM + f4;
    atomicAdd(o + 0, v.x);
    atomicAdd(o + 1, v.y);
    atomicAdd(o + 2, v.z);
    atomicAdd(o + 3, v.w);
}

__global__ void pool_count_kernel(const int* __restrict__ batch, float* __restrict__ pcnt, int Nn) {
    int i = blockIdx.x * blockDim.x + threadIdx.x;
    if (i < Nn) atomicAdd(&pcnt[batch[i]], 1.0f);
}

__global__ void pool_finalize_kernel(const float* __restrict__ psum, const float* __restrict__ pcnt,
                                     float* __restrict__ out, int total) {
    int i = blockIdx.x * blockDim.x + threadIdx.x;
    if (i < total) out[i] = psum[i] / fmaxf(pcnt[i >> 6], 1.0f);   // ODIM == 64
}

// ---------- launch ----------

static inline unsigned nblk(long long n, int t) { return (unsigned)((n + t - 1) / t); }

extern "C" void kernel_launch(void* const* d_in, const int* in_sizes, int n_in,
                              void* d_out, int out_size, void* d_ws, size_t ws_size,
                              hipStream_t stream) {
    const float* x     = (const float*)d_in[0];
    const int*   edges = (const int*)d_in[1];      // [2, E] flattened
    const int*   batch = (const int*)d_in[2];
    const float* W1    = (const float*)d_in[3];
    const float* b1    = (const float*)d_in[4];
    const float* W2    = (const float*)d_in[5];
    const float* b2    = (const float*)d_in[6];
    float* out = (float*)d_out;

    const int N = in_sizes[0] / F_IN;
    const int E = in_sizes[1] / 2;
    const int B = out_size / ODIM;
    const int* src = edges;
    const int* dst = edges + E;

    // workspace layout (floats)
    float* ws    = (float*)d_ws;
    float* dinv  = ws;                                   // N
    float* hbuf  = dinv + N;                             // N*HDIM (h1, then h3)
    float* aggbf = hbuf + (size_t)N * HDIM;              // N*HDIM (agg1, then agg2)
    float* psum  = aggbf + (size_t)N * HDIM;             // B*ODIM
    float* pcnt  = psum + (size_t)B * ODIM;              // B
    (void)ws_size; (void)n_in;

    const int T = 256;

    // degree + normalization
    init_const_kernel<<<nblk(N, T), T, 0, stream>>>(dinv, 1.0f, N);
    degree_kernel<<<nblk(E, T), T, 0, stream>>>(dst, dinv, E);
    dinv_kernel<<<nblk(N, T), T, 0, stream>>>(dinv, N);

    // ---- layer 1 ----
    {   // h1 = x @ W1   [N,64]x[64,128]
        long long waves = (long long)((N + 15) / 16) * (HDIM / 32);
        gemm_wmma_kernel<F_IN, HDIM><<<nblk(waves * 32, T), T, 0, stream>>>(x, W1, hbuf, N);
    }
    selfloop_bias_kernel<<<nblk((long long)N * HDIM, T), T, 0, stream>>>(hbuf, dinv, b1, aggbf, N, 7);
    scatter_kernel<<<nblk((long long)E * (HDIM / 4), T), T, 0, stream>>>(hbuf, src, dst, dinv,
                                                                         aggbf, E, HDIM, 5);
    relu_kernel<<<nblk((long long)N * HDIM, T), T, 0, stream>>>(aggbf, (long long)N * HDIM);

    // ---- layer 2 ----
    {   // h3 = relu(agg1) @ W2   [N,128]x[128,64]  (h3 reuses hbuf)
        long long waves = (long long)((N + 15) / 16) * (ODIM / 32);
        gemm_wmma_kernel<HDIM, ODIM><<<nblk(waves * 32, T), T, 0, stream>>>(aggbf, W2, hbuf, N);
    }
    selfloop_bias_kernel<<<nblk((long long)N * ODIM, T), T, 0, stream>>>(hbuf, dinv, b2, aggbf, N, 6);
    scatter_kernel<<<nblk((long long)E * (ODIM / 4), T), T, 0, stream>>>(hbuf, src, dst, dinv,
                                                                         aggbf, E, ODIM, 4);

    // ---- global mean pool ----
    init_const_kernel<<<nblk(B * ODIM + B, T), T, 0, stream>>>(psum, 0.0f, B * ODIM + B);
    pool_sum_kernel<<<nblk((long long)N * (ODIM / 4), T), T, 0, stream>>>(aggbf, batch, psum, N);
    pool_count_kernel<<<nblk(N, T), T, 0, stream>>>(batch, pcnt, N);
    pool_finalize_kernel<<<nblk(B * ODIM, T), T, 0, stream>>>(psum, pcnt, out, B * ODIM);
}